// GraphAutoencoderTra_51788715655838
// MI455X (gfx1250) — compile-verified
//
#include <hip/hip_runtime.h>
#include <hip/hip_bf16.h>

typedef __attribute__((ext_vector_type(2))) float v2f;
typedef __attribute__((ext_vector_type(8))) float v8f;
typedef __attribute__((ext_vector_type(4))) unsigned int u32x4;
typedef __attribute__((ext_vector_type(8))) int i32x8;
typedef __attribute__((ext_vector_type(4))) int i32x4;

#define ZONE 128
#define RD   128
#define LDIM 32
#define NDIM 96    // RD - LDIM
#define OUTD 100
#define TRROWS 64  // K-rows staged per TDM slab

// ---- Tensor Data Mover: 2D strided tile (global -> LDS), D# per ISA ch.8 ----
// group0: [1:0]=count=1, [63:32]=lds_addr, [120:64]=global_addr, [127:126]=type=2
// group1: [17:16]=data_size(2 -> 4B), [79:48]=tensor_dim0, [111:80]=tensor_dim1,
//         [127:112]=tile_dim0, [143:128]=tile_dim1, [207:160]=tensor_dim0_stride
__device__ __forceinline__ void tdm_load_2d(unsigned int lds_off_bytes,
                                            const float* gptr,
                                            unsigned int tile_w, unsigned int tile_h,
                                            unsigned int tensor_w, unsigned int tensor_h,
                                            unsigned long long row_stride_elems)
{
    unsigned long long ga = (unsigned long long)(uintptr_t)gptr;
    u32x4 g0;
    g0.x = 0x1u;                                    // count = 1 valid descriptor
    g0.y = lds_off_bytes;                           // lds_addr
    g0.z = (unsigned int)ga;                        // global_addr[31:0]
    g0.w = (unsigned int)(ga >> 32) | (2u << 30);   // global_addr[56:32] | type=2
    i32x8 g1 = {0, 0, 0, 0, 0, 0, 0, 0};
    g1[0] = (int)(2u << 16);                                         // data_size = 4B
    g1[1] = (int)((tensor_w & 0xFFFFu) << 16);                       // tensor_dim0 lo16
    g1[2] = (int)((tensor_w >> 16) | ((tensor_h & 0xFFFFu) << 16));  // dim0 hi / dim1 lo
    g1[3] = (int)((tensor_h >> 16) | ((tile_w & 0xFFFFu) << 16));    // dim1 hi / tile_dim0
    g1[4] = (int)(tile_h & 0xFFFFu);                                 // tile_dim1 (tile_dim2=0)
    g1[5] = (int)(unsigned int)row_stride_elems;                     // dim0_stride lo32
    g1[6] = (int)(unsigned int)(row_stride_elems >> 32);             // dim0_stride hi16
    i32x4 z4 = {0, 0, 0, 0};
    i32x8 z8 = {0, 0, 0, 0, 0, 0, 0, 0};
    __builtin_amdgcn_tensor_load_to_lds(g0, g1, z4, z4, z8, 0);
}

// ---------------- column sums of struct_assign ----------------
__global__ void colsum_kernel(const float* __restrict__ sa, float* __restrict__ colsum, int nroad)
{
    int c = threadIdx.x;                       // 0..127
    float acc = 0.f;
    for (int r = blockIdx.x; r < nroad; r += gridDim.x)
        acc += sa[(size_t)r * ZONE + c];
    atomicAdd(&colsum[c], acc);
}

__global__ void recip_kernel(const float* __restrict__ colsum, float* __restrict__ recip)
{
    int c = threadIdx.x;
    recip[c] = 1.0f / (fmaxf(colsum[c] - 1.0f, 0.0f) + 1.0f);
}

// ---------------- raw_feat = concat(length_emb[lf], node_emb[nf]) ----------------
__global__ void rawfeat_kernel(const int* __restrict__ lf, const int* __restrict__ nf,
                               const float* __restrict__ lemb, const float* __restrict__ nemb,
                               float* __restrict__ raw, int nroad)
{
    int r = blockIdx.x;
    int d = threadIdx.x;  // 128
    float v = (d < LDIM) ? lemb[(size_t)lf[r] * LDIM + d]
                         : nemb[(size_t)nf[r] * NDIM + (d - LDIM)];
    raw[(size_t)r * RD + d] = v;
}

// ---------------- struct_inter: scatter-add of sa rows over edges ----------------
__global__ void edge_scatter_kernel(const int* __restrict__ ei, int ne,
                                    const float* __restrict__ sa, const float* __restrict__ recip,
                                    float* __restrict__ inter)
{
    int e = blockIdx.x;
    int c = threadIdx.x;  // 128
    int dst = ei[e];
    int src = ei[ne + e];
    float v = sa[(size_t)src * ZONE + c] * recip[c];
    atomicAdd(&inter[(size_t)dst * ZONE + c], v);
}

// ---------------- C[128,128] += sa[:,z0:z0+16].T @ B over a road chunk ----------------
// Block = 8 waves. Wave 0 DMAs an A slab (TRROWSx16, strided tile) and a B slab
// (TRROWSx128) into LDS with tensor_load_to_lds, waits on TENSORcnt, barrier;
// each wave then runs V_WMMA_F32_16X16X4_F32 over its 16-col n-tile from LDS.
__global__ void __launch_bounds__(256)
wmma_atb_kernel(const float* __restrict__ SA, const float* __restrict__ recip,
                const float* __restrict__ B, float* __restrict__ C,
                int nroad, int rows_per_chunk)
{
    __shared__ __align__(128) float smem[TRROWS * (16 + RD)];
    float* ldsA = smem;                 // LDS offset 0 (single static allocation)
    float* ldsB = smem + TRROWS * 16;   // LDS offset TRROWS*16*4 bytes
    const unsigned int ldsB_off = TRROWS * 16 * sizeof(float);

    const int tid  = threadIdx.x;
    const int wave = tid >> 5;          // 0..7 -> n-tile
    const int lane = tid & 31;
    const int idx  = lane & 15;
    const int half = lane >> 4;
    const int z0 = blockIdx.x * 16;     // m-tile (zone columns)
    const int n0 = wave * 16;           // n-tile within B's 128 columns

    // IR-visible LDS writes (real ds_stores covering the slab). The TDM's LDS
    // writes are invisible to alias analysis; without these, slab reads fold to
    // poison ("never-written memory") and the LDS pipeline is deleted. The
    // workgroup fences in __syncthreads() block forwarding of these zeros.
    for (int i = tid; i < TRROWS * (16 + RD); i += 256)
        smem[i] = 0.0f;

    long r0 = (long)blockIdx.y * rows_per_chunk;
    long r1 = r0 + rows_per_chunk; if (r1 > nroad) r1 = nroad;

    float rA = recip[z0 + idx];
    v8f acc = {};

    for (long rb = r0; rb < r1; rb += TRROWS) {
        int rows = (int)((r1 - rb < TRROWS) ? (r1 - rb) : TRROWS);
        __syncthreads();                // previous slab fully consumed
        if (wave == 0) {
            // A slab: 16-wide column window of struct_assign, row stride 128
            tdm_load_2d(0, SA + rb * ZONE + z0, 16u, (unsigned)rows,
                        (unsigned)ZONE, (unsigned)rows, (unsigned long long)ZONE);
            // B slab: full 128-wide rows
            tdm_load_2d(ldsB_off, B + rb * RD, (unsigned)RD, (unsigned)rows,
                        (unsigned)RD, (unsigned)rows, (unsigned long long)RD);
            __builtin_amdgcn_s_wait_tensorcnt((short)0);
        }
        __syncthreads();                // slab visible to all waves
        int kmax = rows & ~3;
        for (int k = 0; k < kmax; k += 4) {
            int ka = k + 2 * half;
            v2f a, b;
            a.x = ldsA[ka * 16 + idx] * rA;          // A[m][k]   (sa.T tile)
            a.y = ldsA[(ka + 1) * 16 + idx] * rA;    // A[m][k+1]
            b.x = ldsB[ka * RD + n0 + idx];          // B[k][n]
            b.y = ldsB[(ka + 1) * RD + n0 + idx];    // B[k+1][n]
            acc = __builtin_amdgcn_wmma_f32_16x16x4_f32(false, a, false, b,
                                                        (short)0, acc, false, false);
        }
    }
    #pragma unroll
    for (int j = 0; j < 8; ++j) {
        int row = z0 + j + 8 * half;
        atomicAdd(&C[(size_t)row * RD + n0 + idx], acc[j]);
    }
}

// ---------------- generic small GEMM: C = A(MxK) @ B(KxN) [+ addC] [+ bias(col)] ----------------
__global__ void wmma_gemm_kernel(const float* __restrict__ A, int lda,
                                 const float* __restrict__ Bm, int ldb,
                                 const float* __restrict__ addC,
                                 const float* __restrict__ bias,
                                 float* __restrict__ C, int ldc,
                                 int N, int K)
{
    int lane = threadIdx.x;
    int idx = lane & 15, half = lane >> 4;
    int m0 = blockIdx.y * 16;
    int n0 = blockIdx.x * 16;
    int bn = n0 + idx;
    bool inb = bn < N;
    v8f acc = {};
    for (int k = 0; k < K; k += 4) {
        int ka = k + 2 * half;
        v2f a, b;
        a.x = A[(size_t)(m0 + idx) * lda + ka];
        a.y = A[(size_t)(m0 + idx) * lda + ka + 1];
        b.x = inb ? Bm[(size_t)ka * ldb + bn] : 0.f;
        b.y = inb ? Bm[(size_t)(ka + 1) * ldb + bn] : 0.f;
        acc = __builtin_amdgcn_wmma_f32_16x16x4_f32(false, a, false, b, (short)0, acc, false, false);
    }
    if (inb) {
        #pragma unroll
        for (int j = 0; j < 8; ++j) {
            int row = m0 + j + 8 * half;
            float v = acc[j];
            if (addC) v += addC[(size_t)row * ldc + bn];
            if (bias) v += bias[bn];
            C[(size_t)row * ldc + bn] = v;
        }
    }
}

// ---------------- softmax over axis 0 (down each column) of a 128x128 matrix ----------------
__global__ void softmax_col_kernel(const float* __restrict__ L, float* __restrict__ out)
{
    int j = threadIdx.x;   // column
    float m = -3.4e38f;
    for (int i = 0; i < ZONE; ++i) m = fmaxf(m, L[i * ZONE + j]);
    float s = 0.f;
    for (int i = 0; i < ZONE; ++i) s += __expf(L[i * ZONE + j] - m);
    float inv = 1.0f / s;
    for (int i = 0; i < ZONE; ++i) out[i * ZONE + j] = __expf(L[i * ZONE + j] - m) * inv;
}

__global__ void transpose128_kernel(const float* __restrict__ A, float* __restrict__ T)
{
    int i = blockIdx.x, j = threadIdx.x;
    T[j * ZONE + i] = A[i * ZONE + j];
}

// ---------------- fused X = raw_feat@W0 + sa@Wc + lin_b  (ROAD x 100) ----------------
__global__ void x_kernel(const float* __restrict__ raw,
                         const float* __restrict__ SA, const float* __restrict__ recip,
                         const float* __restrict__ W0, const float* __restrict__ Wc,
                         const float* __restrict__ lb,
                         float* __restrict__ X, int nroad)
{
    int lane = threadIdx.x;
    int idx = lane & 15, half = lane >> 4;
    long r0 = (long)blockIdx.x * 16;
    int n0 = blockIdx.y * 16;
    int bn = n0 + idx;
    bool inb = bn < OUTD;
    long row = r0 + idx; if (row >= nroad) row = nroad - 1;
    v8f acc = {};
    for (int k = 0; k < RD; k += 4) {
        int ka = k + 2 * half;
        v2f a, b;
        a.x = raw[row * RD + ka];
        a.y = raw[row * RD + ka + 1];
        b.x = inb ? W0[(size_t)ka * OUTD + bn] : 0.f;
        b.y = inb ? W0[(size_t)(ka + 1) * OUTD + bn] : 0.f;
        acc = __builtin_amdgcn_wmma_f32_16x16x4_f32(false, a, false, b, (short)0, acc, false, false);
    }
    for (int k = 0; k < ZONE; k += 4) {
        int ka = k + 2 * half;
        v2f a, b;
        a.x = SA[row * ZONE + ka] * recip[ka];
        a.y = SA[row * ZONE + ka + 1] * recip[ka + 1];
        b.x = inb ? Wc[(size_t)ka * OUTD + bn] : 0.f;
        b.y = inb ? Wc[(size_t)(ka + 1) * OUTD + bn] : 0.f;
        acc = __builtin_amdgcn_wmma_f32_16x16x4_f32(false, a, false, b, (short)0, acc, false, false);
    }
    if (inb) {
        float bias = lb[bn];
        #pragma unroll
        for (int j = 0; j < 8; ++j) {
            long rr = r0 + j + 8 * half;
            if (rr < nroad) X[rr * OUTD + bn] = acc[j] + bias;
        }
    }
}

// ---------------- pred_edge[s] = dot(X[a], X[b]) ; one wave32 per edge ----------------
__global__ void pred_edge_kernel(const int* __restrict__ se, int ns,
                                 const float* __restrict__ X, float* __restrict__ out)
{
    int wave = threadIdx.y;
    int lane = threadIdx.x;
    long s = (long)blockIdx.x * blockDim.y + wave;
    if (s >= ns) return;
    long a = se[s];
    long b = se[ns + s];
    const float* xa = X + a * OUTD;
    const float* xb = X + b * OUTD;
    float acc = 0.f;
    for (int c = lane; c < OUTD; c += 32)
        acc += xa[c] * xb[c];
    #pragma unroll
    for (int off = 16; off > 0; off >>= 1)
        acc += __shfl_down(acc, off, 32);
    if (lane == 0) out[s] = acc;
}

extern "C" void kernel_launch(void* const* d_in, const int* in_sizes, int n_in,
                              void* d_out, int out_size, void* d_ws, size_t ws_size,
                              hipStream_t stream)
{
    const int*   length_feature = (const int*)d_in[0];
    const int*   node_feature   = (const int*)d_in[1];
    const int*   edge_index     = (const int*)d_in[2];
    const float* struct_assign  = (const float*)d_in[3];
    const int*   s_edge         = (const int*)d_in[4];
    const float* node_emb       = (const float*)d_in[5];
    const float* length_emb     = (const float*)d_in[6];
    const float* gcn_w          = (const float*)d_in[7];
    const float* gcn_b          = (const float*)d_in[8];
    const float* lin_w          = (const float*)d_in[9];
    const float* lin_b          = (const float*)d_in[10];

    const int nroad = in_sizes[0];
    const int ne    = in_sizes[2] / 2;
    const int ns    = in_sizes[4] / 2;

    float* pred = (float*)d_out;        // [ns]
    float* X    = pred + ns;            // [nroad * 100]

    char* ws = (char*)d_ws;
    size_t off = 0;
    auto alloc = [&](size_t bytes) -> void* {
        void* p = ws + off;
        off = (off + bytes + 255) & ~(size_t)255;
        return p;
    };
    float* raw    = (float*)alloc((size_t)nroad * RD * sizeof(float));
    float* inter  = (float*)alloc((size_t)nroad * ZONE * sizeof(float));
    float* colsum = (float*)alloc(ZONE * sizeof(float));
    float* recip  = (float*)alloc(ZONE * sizeof(float));
    float* semb   = (float*)alloc(ZONE * RD * sizeof(float));    // struct_emb
    float* sadj   = (float*)alloc(ZONE * ZONE * sizeof(float));  // struct_adj
    float* T1     = (float*)alloc(ZONE * ZONE * sizeof(float));
    float* Lg     = (float*)alloc(ZONE * ZONE * sizeof(float));  // gcn logits
    float* fa     = (float*)alloc(ZONE * ZONE * sizeof(float));  // fnc_assign
    float* faT    = (float*)alloc(ZONE * ZONE * sizeof(float));
    float* femb   = (float*)alloc(ZONE * RD * sizeof(float));    // fnc_emb
    float* U      = (float*)alloc(ZONE * OUTD * sizeof(float));  // fnc_emb @ W2
    float* Wc1    = (float*)alloc(ZONE * OUTD * sizeof(float));  // struct_emb @ W1
    float* Wc     = (float*)alloc(ZONE * OUTD * sizeof(float));  // folded weight
    (void)ws_size; (void)n_in; (void)out_size;

    (void)hipMemsetAsync(colsum, 0, ZONE * sizeof(float), stream);
    (void)hipMemsetAsync(semb, 0, ZONE * RD * sizeof(float), stream);
    (void)hipMemsetAsync(sadj, 0, ZONE * ZONE * sizeof(float), stream);
    (void)hipMemsetAsync(inter, 0, (size_t)nroad * ZONE * sizeof(float), stream);

    colsum_kernel<<<512, ZONE, 0, stream>>>(struct_assign, colsum, nroad);
    recip_kernel<<<1, ZONE, 0, stream>>>(colsum, recip);
    rawfeat_kernel<<<nroad, RD, 0, stream>>>(length_feature, node_feature,
                                             length_emb, node_emb, raw, nroad);

    const int CHUNK = 2048;
    const int nchunks = (nroad + CHUNK - 1) / CHUNK;
    // struct_emb = sa.T @ raw_feat  (TDM-staged LDS, 8 waves/block)
    wmma_atb_kernel<<<dim3(8, nchunks), 256, 0, stream>>>(struct_assign, recip, raw,
                                                          semb, nroad, CHUNK);
    // struct_inter = segment_sum(sa[src], dst)
    edge_scatter_kernel<<<ne, ZONE, 0, stream>>>(edge_index, ne, struct_assign, recip, inter);
    // struct_adj = sa.T @ struct_inter
    wmma_atb_kernel<<<dim3(8, nchunks), 256, 0, stream>>>(struct_assign, recip, inter,
                                                          sadj, nroad, CHUNK);

    const float* W0 = lin_w;                  // rows 0..127
    const float* W1 = lin_w + 1 * RD * OUTD;  // rows 128..255
    const float* W2 = lin_w + 2 * RD * OUTD;  // rows 256..383

    // fnc_assign = softmax_col(struct_adj @ (struct_emb @ gcn_w) + gcn_b)
    wmma_gemm_kernel<<<dim3(8, 8), 32, 0, stream>>>(semb, RD, gcn_w, ZONE,
                                                    nullptr, nullptr, T1, ZONE, ZONE, RD);
    wmma_gemm_kernel<<<dim3(8, 8), 32, 0, stream>>>(sadj, ZONE, T1, ZONE,
                                                    nullptr, gcn_b, Lg, ZONE, ZONE, ZONE);
    softmax_col_kernel<<<1, ZONE, 0, stream>>>(Lg, fa);
    transpose128_kernel<<<ZONE, ZONE, 0, stream>>>(fa, faT);
    // fnc_emb = fnc_assign.T @ struct_emb
    wmma_gemm_kernel<<<dim3(8, 8), 32, 0, stream>>>(faT, ZONE, semb, RD,
                                                    nullptr, nullptr, femb, RD, RD, ZONE);
    // Wc = struct_emb @ W1 + fnc_assign @ (fnc_emb @ W2)
    wmma_gemm_kernel<<<dim3(7, 8), 32, 0, stream>>>(femb, RD, W2, OUTD,
                                                    nullptr, nullptr, U, OUTD, OUTD, RD);
    wmma_gemm_kernel<<<dim3(7, 8), 32, 0, stream>>>(semb, RD, W1, OUTD,
                                                    nullptr, nullptr, Wc1, OUTD, OUTD, RD);
    wmma_gemm_kernel<<<dim3(7, 8), 32, 0, stream>>>(fa, ZONE, U, OUTD,
                                                    Wc1, nullptr, Wc, OUTD, OUTD, ZONE);

    // X = raw_feat @ W0 + sa @ Wc + lin_b
    const int mtiles = (nroad + 15) / 16;
    x_kernel<<<dim3(mtiles, 7), 32, 0, stream>>>(raw, struct_assign, recip,
                                                 W0, Wc, lin_b, X, nroad);

    // pred_edge = rowwise dot of gathered X rows
    pred_edge_kernel<<<(ns + 7) / 8, dim3(32, 8), 0, stream>>>(s_edge, ns, X, pred);
}